// GaussianOrientedBboxIou_35416300323280
// MI455X (gfx1250) — compile-verified
//
#include <hip/hip_runtime.h>
#include <hip/hip_bf16.h>
#include <stdint.h>

#define TILE 32
#define FSTRIDE 8   // floats per box-feature record (6 used, padded to 32B)

#define AS1 __attribute__((address_space(1)))
#define AS3 __attribute__((address_space(3)))

typedef __attribute__((ext_vector_type(4))) int   v4i_t;
typedef __attribute__((ext_vector_type(4))) float v4f_t;

// ---------------- gfx1250 async global->LDS helpers ----------------
__device__ __forceinline__ void async_b128_to_lds(const float* g, float* l) {
#if __has_builtin(__builtin_amdgcn_global_load_async_to_lds_b128)
    __builtin_amdgcn_global_load_async_to_lds_b128(
        (AS1 v4i_t*)(uintptr_t)g, (AS3 v4i_t*)l, 0, 0);
#else
    unsigned lds_off = (unsigned)(uintptr_t)(AS3 void*)l;
    asm volatile("global_load_async_to_lds_b128 %0, %1, off"
                 :: "v"(lds_off), "v"(g) : "memory");
#endif
}

__device__ __forceinline__ void wait_async0() {
#if __has_builtin(__builtin_amdgcn_s_wait_asynccnt)
    __builtin_amdgcn_s_wait_asynccnt(0);
#else
    asm volatile("s_wait_asynccnt 0" ::: "memory");
#endif
}

// ---------------- per-box feature extraction ----------------
// dst[0..5] = ux, uy, s00, s01, s11, 0.25*log(det(sigma))
__device__ __forceinline__ void compute_feat(const float* __restrict__ bb8,
                                             float angle,
                                             float* dst) {
    float4 c01 = ((const float4*)bb8)[0];
    float4 c23 = ((const float4*)bb8)[1];
    float x0 = c01.x, y0 = c01.y, x1 = c01.z, y1 = c01.w;
    float x2 = c23.x, y2 = c23.y, x3 = c23.z, y3 = c23.w;

    float cx = (x0 + x1 + x2 + x3) * 0.25f;
    float cy = (y0 + y1 + y2 + y3) * 0.25f;

    float sn, cs;
    sincosf(angle, &sn, &cs);

    // pts = (p - c) @ R(-angle): px = dx*cs - dy*sn ; py = dx*sn + dy*cs
    float dx0 = x0 - cx, dy0 = y0 - cy;
    float dx1 = x1 - cx, dy1 = y1 - cy;
    float dx2 = x2 - cx, dy2 = y2 - cy;
    float dx3 = x3 - cx, dy3 = y3 - cy;

    float px0 = dx0 * cs - dy0 * sn, py0 = dx0 * sn + dy0 * cs;
    float px1 = dx1 * cs - dy1 * sn, py1 = dx1 * sn + dy1 * cs;
    float px2 = dx2 * cs - dy2 * sn, py2 = dx2 * sn + dy2 * cs;
    float px3 = dx3 * cs - dy3 * sn, py3 = dx3 * sn + dy3 * cs;

    float mxx = fmaxf(fmaxf(px0, px1), fmaxf(px2, px3));
    float mnx = fminf(fminf(px0, px1), fminf(px2, px3));
    float mxy = fmaxf(fmaxf(py0, py1), fmaxf(py2, py3));
    float mny = fminf(fminf(py0, py1), fminf(py2, py3));

    float wx = (mxx - mnx) * 0.5f;   // size/2 along rotated x
    float wy = (mxy - mny) * 0.5f;   // size/2 along rotated y

    // sigma = R(a) diag(wx,wy) R(a)^T
    float s00 = cs * cs * wx + sn * sn * wy;
    float s01 = cs * sn * (wx - wy);
    float s11 = sn * sn * wx + cs * cs * wy;
    float det_s = s00 * s11 - s01 * s01;

    dst[0] = cx;  dst[1] = cy;
    dst[2] = s00; dst[3] = s01; dst[4] = s11;
    dst[5] = 0.25f * logf(det_s);   // pairwise uses 0.5*log(det) - li - lj
}

// ---------------- pairwise IoU (Hellinger), fast-math pair body ----------------
__device__ __forceinline__ float pair_iou(float uxi, float uyi, float ai,
                                          float bi, float di, float li,
                                          const float* fj) {
    float dx = uxi - fj[0];
    float dy = uyi - fj[1];
    float a  = (ai + fj[2]) * 0.5f;
    float bb = (bi + fj[3]) * 0.5f;
    float d  = (di + fj[4]) * 0.5f;
    float det = __builtin_fmaf(a, d, -(bb * bb));
    float inv = (det == 0.0f) ? 0.0f : __builtin_amdgcn_rcpf(det);
    // Bd quadratic term: reference uses diag of the inverse only
    float t  = (d * dx * dx + a * dy * dy) * inv * 0.125f;
    float Bd = t + 0.5f * __logf(det) - li - fj[5];
    float Bc = __expf(-Bd);
    float Hd = __builtin_sqrtf(fmaxf(1.0f - Bc, 1e-12f));
    return 1.0f - Hd;
}

// ---------------- phase 1: per-box features ----------------
__global__ __launch_bounds__(256)
void box_feat_kernel(const float* __restrict__ bboxes,
                     const float* __restrict__ angles,
                     float* __restrict__ feat, int M) {
    int idx = blockIdx.x * blockDim.x + threadIdx.x;
    if (idx >= M) return;
    float f[6];
    compute_feat(bboxes + (size_t)idx * 8, angles[idx], f);
    float4* o = (float4*)(feat + (size_t)idx * FSTRIDE);
    o[0] = make_float4(f[0], f[1], f[2], f[3]);
    o[1] = make_float4(f[4], f[5], 0.0f, 0.0f);
}

// ---------------- phase 2: pairwise tile kernel ----------------
__global__ __launch_bounds__(256)
void pair_kernel(const float* __restrict__ feat,
                 float* __restrict__ out, int N) {
    __shared__ __align__(16) float sRow[TILE * FSTRIDE];
    __shared__ __align__(16) float sCol[TILE * FSTRIDE];

    const int b  = blockIdx.z;
    const int i0 = blockIdx.y * TILE;
    const int j0 = blockIdx.x * TILE;
    const float* fb = feat + (size_t)b * N * FSTRIDE;

    const int t = threadIdx.x;

    // Stage 2x 1KB of per-box features into LDS via async b128 copies.
    // 64 x 16-byte chunks per side; threads 0..127 issue one each.
    if (t < 128) {
        int side = t >> 6;                      // 0 = row tile, 1 = col tile
        int k    = t & 63;                      // 16-byte chunk id (2 per box)
        int box  = (side ? j0 : i0) + (k >> 1);
        const float* g = fb + (size_t)box * FSTRIDE + (k & 1) * 4;
        float* l = (side ? sCol : sRow) + k * 4;
        async_b128_to_lds(g, l);
    }
    wait_async0();
    __syncthreads();

    const int ty = t >> 3;        // 0..31  : row within tile
    const int tx = t & 7;         // 0..7   : 4-wide column strip
    const int i  = i0 + ty;

    const float* fi = sRow + ty * FSTRIDE;
    const float uxi = fi[0], uyi = fi[1], ai = fi[2];
    const float bi  = fi[3], di  = fi[4], li = fi[5];
    const float* fj = sCol + (tx * 4) * FSTRIDE;

    v4f_t r;
    r.x = pair_iou(uxi, uyi, ai, bi, di, li, fj + 0 * FSTRIDE);
    r.y = pair_iou(uxi, uyi, ai, bi, di, li, fj + 1 * FSTRIDE);
    r.z = pair_iou(uxi, uyi, ai, bi, di, li, fj + 2 * FSTRIDE);
    r.w = pair_iou(uxi, uyi, ai, bi, di, li, fj + 3 * FSTRIDE);

    // write-only 67MB stream: non-temporal store
    v4f_t* dst = (v4f_t*)(out + (size_t)b * N * N + (size_t)i * N + (j0 + tx * 4));
    __builtin_nontemporal_store(r, dst);
}

// ---------------- fallback: fused (no workspace) ----------------
__global__ __launch_bounds__(256)
void pair_fused_kernel(const float* __restrict__ bboxes,
                       const float* __restrict__ angles,
                       float* __restrict__ out, int N) {
    __shared__ __align__(16) float sRow[TILE * FSTRIDE];
    __shared__ __align__(16) float sCol[TILE * FSTRIDE];

    const int b  = blockIdx.z;
    const int i0 = blockIdx.y * TILE;
    const int j0 = blockIdx.x * TILE;
    const int t  = threadIdx.x;

    if (t < 64) {
        int side = t >> 5;
        int k    = t & 31;
        size_t gi = (size_t)b * N + (side ? j0 : i0) + k;
        compute_feat(bboxes + gi * 8, angles[gi],
                     (side ? sCol : sRow) + k * FSTRIDE);
    }
    __syncthreads();

    const int ty = t >> 3;
    const int tx = t & 7;
    const int i  = i0 + ty;

    const float* fi = sRow + ty * FSTRIDE;
    const float uxi = fi[0], uyi = fi[1], ai = fi[2];
    const float bi  = fi[3], di  = fi[4], li = fi[5];
    const float* fj = sCol + (tx * 4) * FSTRIDE;

    v4f_t r;
    r.x = pair_iou(uxi, uyi, ai, bi, di, li, fj + 0 * FSTRIDE);
    r.y = pair_iou(uxi, uyi, ai, bi, di, li, fj + 1 * FSTRIDE);
    r.z = pair_iou(uxi, uyi, ai, bi, di, li, fj + 2 * FSTRIDE);
    r.w = pair_iou(uxi, uyi, ai, bi, di, li, fj + 3 * FSTRIDE);

    v4f_t* dst = (v4f_t*)(out + (size_t)b * N * N + (size_t)i * N + (j0 + tx * 4));
    __builtin_nontemporal_store(r, dst);
}

extern "C" void kernel_launch(void* const* d_in, const int* in_sizes, int n_in,
                              void* d_out, int out_size, void* d_ws, size_t ws_size,
                              hipStream_t stream) {
    const float* bboxes = (const float*)d_in[0];   // (B, N, 4, 2) f32
    const float* angles = (const float*)d_in[1];   // (B, N)       f32
    float* out = (float*)d_out;                    // (B, N, N)    f32

    const int M = in_sizes[1];          // B*N boxes
    const int N = out_size / M;         // boxes per batch (2048)
    const int B = M / N;                // batches (4)

    dim3 grid(N / TILE, N / TILE, B);

    const size_t need = (size_t)M * FSTRIDE * sizeof(float);
    if (ws_size >= need) {
        float* feat = (float*)d_ws;
        box_feat_kernel<<<(M + 255) / 256, 256, 0, stream>>>(bboxes, angles, feat, M);
        pair_kernel<<<grid, 256, 0, stream>>>(feat, out, N);
    } else {
        pair_fused_kernel<<<grid, 256, 0, stream>>>(bboxes, angles, out, N);
    }
}